// HaarWavelet_43662637531199
// MI455X (gfx1250) — compile-verified
//
#include <hip/hip_runtime.h>

typedef float v2f __attribute__((ext_vector_type(2)));
typedef float v8f __attribute__((ext_vector_type(8)));

// One wave32 processes one 16x16 input tile of a 512x512 plane:
//   D = X * L^T  via 4x V_WMMA_F32_16X16X4_F32  (column butterfly, full fp32)
//   row butterfly done on the 8 accumulator VGPRs (row pairs share a lane)
// where L is the 16x16 Haar pair-butterfly matrix:
//   rows 0..7 : L[i][2i] = 1, L[i][2i+1] = 1      (pair sums)
//   rows 8..15: L[8+i][2i] = 1, L[8+i][2i+1] = -1 (pair diffs)
__global__ __launch_bounds__(256) void haar_wmma_kernel(
    const float* __restrict__ in, float* __restrict__ out) {
  const int tid  = threadIdx.x;
  const int lane = tid & 31;
  const int r    = lane & 15;  // row within tile (per half-wave); also N col
  const int h    = lane >> 4;  // which half of the wave

  // global wave id == tile id; 1024 tiles (32x32) per 512x512 plane
  const int gw    = blockIdx.x * 8 + (tid >> 5);
  const int plane = gw >> 10;
  const int tile  = gw & 1023;
  const int R0    = (tile >> 5) << 4;  // tile row origin
  const int C0    = (tile & 31) << 4;  // tile col origin

  // Each lane loads 8 contiguous floats of row r: cols 8h .. 8h+7 (two b128s)
  const float* rowp =
      in + ((size_t)plane * 512 + (size_t)(R0 + r)) * 512 + (C0 + 8 * h);
  const float4 vlo = *(const float4*)(rowp);
  const float4 vhi = *(const float4*)(rowp + 4);
  const float x0 = vlo.x, x1 = vlo.y, x2 = vlo.z, x3 = vlo.w;
  const float x4 = vhi.x, x5 = vhi.y, x6 = vhi.z, x7 = vhi.w;

  // Cross-half swaps (lane <-> lane^16) to build the fp32 A-matrix 16x4 layout:
  // VGPR0: lanes0-15 K=4k, lanes16-31 K=4k+2 ; VGPR1: K=4k+1 / K=4k+3
  const float s0 = __shfl_xor(x0, 16, 32), s1 = __shfl_xor(x1, 16, 32);
  const float s2 = __shfl_xor(x2, 16, 32), s3 = __shfl_xor(x3, 16, 32);
  const float s4 = __shfl_xor(x4, 16, 32), s5 = __shfl_xor(x5, 16, 32);
  const float s6 = __shfl_xor(x6, 16, 32), s7 = __shfl_xor(x7, 16, 32);

  const v2f a0 = {h ? s2 : x0, h ? s3 : x1};  // K slice 0..3
  const v2f a1 = {h ? s6 : x4, h ? s7 : x5};  // K slice 4..7
  const v2f a2 = {h ? x2 : s0, h ? x3 : s1};  // K slice 8..11
  const v2f a3 = {h ? x6 : s4, h ? x7 : s5};  // K slice 12..15

  // B = L^T K-slices, built branchlessly (no lane-divergent CFG; EXEC stays
  // all-ones through the WMMAs).  fp32 B 4x16 layout mirrors A: VGPR0 holds
  // K=4k (lanes 0-15) / K=4k+2 (lanes 16-31); VGPR1 K=4k+1 / K=4k+3.
  // Lt[c][n] nonzero iff c>>1 == (n&7); even c -> +1, odd c -> +1 (n<8) / -1.
  const int   n7  = r & 7;
  const float odd = (r < 8) ? 1.0f : -1.0f;  // sign of odd-column entries
  const float m0  = (n7 == 0 + h) ? 1.0f : 0.0f;
  const float m1  = (n7 == 2 + h) ? 1.0f : 0.0f;
  const float m2  = (n7 == 4 + h) ? 1.0f : 0.0f;
  const float m3  = (n7 == 6 + h) ? 1.0f : 0.0f;
  const v2f b0 = {m0, m0 * odd};
  const v2f b1 = {m1, m1 * odd};
  const v2f b2 = {m2, m2 * odd};
  const v2f b3 = {m3, m3 * odd};

  v8f acc = {0.f, 0.f, 0.f, 0.f, 0.f, 0.f, 0.f, 0.f};
  acc = __builtin_amdgcn_wmma_f32_16x16x4_f32(false, a0, false, b0, (short)0, acc, false, false);
  acc = __builtin_amdgcn_wmma_f32_16x16x4_f32(false, a1, false, b1, (short)0, acc, false, false);
  acc = __builtin_amdgcn_wmma_f32_16x16x4_f32(false, a2, false, b2, (short)0, acc, false, false);
  acc = __builtin_amdgcn_wmma_f32_16x16x4_f32(false, a3, false, b3, (short)0, acc, false, false);

  // D layout: VGPR v holds row M=v (lanes 0-15) and M=v+8 (lanes 16-31),
  // column N = lane%16.  N<8 = column-sums, N>=8 = column-diffs.
  // Row butterfly: combine adjacent accumulator VGPRs (rows 2q, 2q+1).
  float sums[4], difs[4];
#pragma unroll
  for (int q = 0; q < 4; ++q) {
    sums[q] = acc[2 * q] + acc[2 * q + 1];  // N<8: a+b+c+d   N>=8: 2*dv
    difs[q] = acc[2 * q] - acc[2 * q + 1];  // N<8: 2*dh      N>=8: dd
  }

  const float sscale = (r < 8) ? 0.25f : 0.5f;  // pooled | dv
  const float dscale = (r < 8) ? 0.5f  : 1.0f;  // dh     | dd

  // Output: (16, 128, 256, 256); sub-band channel groups of 32.
  const int bidx   = plane >> 5;   // batch
  const int ch     = plane & 31;   // input channel
  const int j      = r & 7;        // output col within tile
  const int colSel = r >> 3;       // 0: {pooled,dh}-lanes, 1: {dv,dd}-lanes
  const int planeSum  = bidx * 128 + ch + (colSel ? 64 : 0);   // pooled / dv
  const int planeDiff = bidx * 128 + ch + (colSel ? 96 : 32);  // dh / dd
  const int oh0 = (R0 >> 1) + 4 * h;  // output row base (q adds 0..3)
  const int ow  = (C0 >> 1) + j;

  float* ps = out + ((size_t)planeSum  * 256 + oh0) * 256 + ow;
  float* pd = out + ((size_t)planeDiff * 256 + oh0) * 256 + ow;
#pragma unroll
  for (int q = 0; q < 4; ++q) {
    ps[q * 256] = sums[q] * sscale;
    pd[q * 256] = difs[q] * dscale;
  }
}

extern "C" void kernel_launch(void* const* d_in, const int* in_sizes, int n_in,
                              void* d_out, int out_size, void* d_ws, size_t ws_size,
                              hipStream_t stream) {
  (void)in_sizes; (void)n_in; (void)out_size; (void)d_ws; (void)ws_size;
  const float* x = (const float*)d_in[0];
  float* y = (float*)d_out;
  // 512 planes * 1024 tiles = 524288 waves; 8 waves (256 threads) per block.
  dim3 grid(65536), block(256);
  hipLaunchKernelGGL(haar_wmma_kernel, grid, block, 0, stream, x, y);
}